// Decoder_89309549953746
// MI455X (gfx1250) — compile-verified
//
#include <hip/hip_runtime.h>
#include <stdint.h>

typedef unsigned int uint;
typedef unsigned long long ull;

#define NB 4
#define TOPK 1000
#define MAXDET 100
#define KP 1008
#define NTILE 63
#define MAXCAND 2048
#define NMS_THR_C 0.5f
#define MIN_SCORE_C 0.5f
#define MIN_VOL_C 10.0f

typedef float v2f __attribute__((ext_vector_type(2)));
typedef float v8f __attribute__((ext_vector_type(8)));

// ---- workspace layout (32-bit word offsets). First ZWORDS words are zeroed each call.
enum : uint32_t {
  OFF_HIST1 = 0,                          // NB*2048
  OFF_HIST2 = OFF_HIST1 + NB * 2048,      // NB*2048
  OFF_CCNT  = OFF_HIST2 + NB * 2048,      // 16 (NB used)
  OFF_CKEY  = OFF_CCNT + 16,              // NB*MAXCAND
  OFF_SMASK = OFF_CKEY + NB * MAXCAND,    // NB*KP*32
  ZWORDS    = OFF_SMASK + NB * KP * 32,
  OFF_CIDX  = ZWORDS,                     // NB*MAXCAND
  OFF_SEL   = OFF_CIDX + NB * MAXCAND,    // NB*8
  OFF_TSC   = OFF_SEL + NB * 8,           // NB*TOPK (float)
  OFF_TIX   = OFF_TSC + NB * TOPK,        // NB*TOPK
  OFF_BOX   = OFF_TIX + NB * TOPK,        // NB*KP*6 (float)
  OFF_VOL   = OFF_BOX + NB * KP * 6,      // NB*KP (float)
  OFF_VALID = OFF_VOL + NB * KP           // NB*KP
};

static __device__ __forceinline__ uint fkey(float f) {
  uint b = __float_as_uint(f);
  return (b & 0x80000000u) ? ~b : (b | 0x80000000u);
}
static __device__ __forceinline__ float fkey_inv(uint k) {
  uint b = (k & 0x80000000u) ? (k & 0x7FFFFFFFu) : ~k;
  return __uint_as_float(b);
}
static __device__ __forceinline__ uint lds_off(const void* p) {
  // flat LDS address: shared aperture base in the high bits, LDS offset in the low 32 bits  (ISA 10.2 aperture table)
  return (uint)(uintptr_t)p;
}
static __device__ __forceinline__ void async_ld_b128(uint ldsoff, const void* gptr) {
  asm volatile("global_load_async_to_lds_b128 %0, %1, off"
               :: "v"(ldsoff), "v"((ull)(uintptr_t)gptr) : "memory");
}

__global__ void k_zero(uint* __restrict__ ws, int n) {
  for (int i = blockIdx.x * blockDim.x + threadIdx.x; i < n; i += gridDim.x * blockDim.x)
    ws[i] = 0u;
}

// ---- pass 1 histogram (monotonic key >> 21), streamed via async loads to LDS ----
__global__ void k_hist1(const float* __restrict__ cls, uint* __restrict__ hist, int N) {
  const int b = blockIdx.y;
  const float4* src = (const float4*)(cls + (size_t)b * N);
  const long long NV4 = N >> 2;
  __shared__ uint hs[4][2048];
  __shared__ float4 stage[8][2][32];
  const int tid = threadIdx.x;
  for (int h = tid; h < 4 * 2048; h += 256) ((uint*)hs)[h] = 0u;
  __syncthreads();
  const int wave = tid >> 5, lane = tid & 31, sub = tid & 3;
  const long long base = (long long)blockIdx.x * 256 + tid;
  const long long stride = (long long)gridDim.x * 256;
  const long long wstart = base - lane;
  const int niter = (wstart < NV4) ? (int)((NV4 - 1 - wstart) / stride) + 1 : 0;
  const uint l0 = lds_off(&stage[wave][0][lane]);
  const uint l1 = lds_off(&stage[wave][1][lane]);
  if (niter > 0) {
    long long i0 = base < NV4 ? base : NV4 - 1;
    async_ld_b128(l0, src + i0);
  }
  for (int it = 0; it < niter; ++it) {
    if (it + 1 < niter) {
      long long inx = base + (long long)(it + 1) * stride;
      if (inx >= NV4) inx = NV4 - 1;
      async_ld_b128(((it + 1) & 1) ? l1 : l0, src + inx);
      asm volatile("s_wait_asynccnt 1" ::: "memory");
    } else {
      asm volatile("s_wait_asynccnt 0" ::: "memory");
    }
    float4 v = stage[wave][it & 1][lane];
    if (base + (long long)it * stride < NV4) {
      atomicAdd(&hs[sub][fkey(v.x) >> 21], 1u);
      atomicAdd(&hs[sub][fkey(v.y) >> 21], 1u);
      atomicAdd(&hs[sub][fkey(v.z) >> 21], 1u);
      atomicAdd(&hs[sub][fkey(v.w) >> 21], 1u);
    }
  }
  __syncthreads();
  for (int h = tid; h < 2048; h += 256) {
    uint s = hs[0][h] + hs[1][h] + hs[2][h] + hs[3][h];
    if (s) atomicAdd(&hist[(size_t)b * 2048 + h], s);
  }
}

__global__ void k_select1(const uint* __restrict__ hist, uint* __restrict__ sel) {
  const int b = blockIdx.x;
  if (threadIdx.x != 0) return;
  uint cum = 0; int b1 = 0;
  for (int h = 2047; h >= 0; --h) {
    uint c = hist[(size_t)b * 2048 + h];
    if (cum + c >= TOPK || h == 0) { b1 = h; break; }
    cum += c;
  }
  sel[b * 8 + 0] = (uint)b1;
  sel[b * 8 + 1] = cum;
}

// ---- pass 2 histogram: key bits [20:10] of elements whose key>>21 == b1 ----
__global__ void k_hist2(const float* __restrict__ cls, uint* __restrict__ hist,
                        const uint* __restrict__ sel, int N) {
  const int b = blockIdx.y;
  const uint b1 = sel[b * 8 + 0];
  const float4* src = (const float4*)(cls + (size_t)b * N);
  const long long NV4 = N >> 2;
  __shared__ uint hs[4][2048];
  __shared__ float4 stage[8][2][32];
  const int tid = threadIdx.x;
  for (int h = tid; h < 4 * 2048; h += 256) ((uint*)hs)[h] = 0u;
  __syncthreads();
  const int wave = tid >> 5, lane = tid & 31, sub = tid & 3;
  const long long base = (long long)blockIdx.x * 256 + tid;
  const long long stride = (long long)gridDim.x * 256;
  const long long wstart = base - lane;
  const int niter = (wstart < NV4) ? (int)((NV4 - 1 - wstart) / stride) + 1 : 0;
  const uint l0 = lds_off(&stage[wave][0][lane]);
  const uint l1 = lds_off(&stage[wave][1][lane]);
  if (niter > 0) {
    long long i0 = base < NV4 ? base : NV4 - 1;
    async_ld_b128(l0, src + i0);
  }
  for (int it = 0; it < niter; ++it) {
    if (it + 1 < niter) {
      long long inx = base + (long long)(it + 1) * stride;
      if (inx >= NV4) inx = NV4 - 1;
      async_ld_b128(((it + 1) & 1) ? l1 : l0, src + inx);
      asm volatile("s_wait_asynccnt 1" ::: "memory");
    } else {
      asm volatile("s_wait_asynccnt 0" ::: "memory");
    }
    float4 v = stage[wave][it & 1][lane];
    if (base + (long long)it * stride < NV4) {
      float f[4] = {v.x, v.y, v.z, v.w};
      #pragma unroll
      for (int c = 0; c < 4; ++c) {
        uint k = fkey(f[c]);
        if ((k >> 21) == b1) atomicAdd(&hs[sub][(k >> 10) & 2047u], 1u);
      }
    }
  }
  __syncthreads();
  for (int h = tid; h < 2048; h += 256) {
    uint s = hs[0][h] + hs[1][h] + hs[2][h] + hs[3][h];
    if (s) atomicAdd(&hist[(size_t)b * 2048 + h], s);
  }
}

__global__ void k_select2(const uint* __restrict__ hist, uint* __restrict__ sel) {
  const int b = blockIdx.x;
  if (threadIdx.x != 0) return;
  uint cum = sel[b * 8 + 1];
  const uint b1 = sel[b * 8 + 0];
  int b2 = 0;
  for (int h = 2047; h >= 0; --h) {
    uint c = hist[(size_t)b * 2048 + h];
    if (cum + c >= TOPK || h == 0) { b2 = h; break; }
    cum += c;
  }
  sel[b * 8 + 2] = (b1 << 11) | (uint)b2;   // 22-bit threshold
}

__global__ void k_compact(const float* __restrict__ cls, const uint* __restrict__ sel,
                          uint* __restrict__ ccnt, uint* __restrict__ ckey,
                          uint* __restrict__ cidx, int N) {
  const int b = blockIdx.y;
  const uint thr = sel[b * 8 + 2];
  const float4* src = (const float4*)(cls + (size_t)b * N);
  const long long NV4 = N >> 2;
  for (long long i = (long long)blockIdx.x * blockDim.x + threadIdx.x; i < NV4;
       i += (long long)gridDim.x * blockDim.x) {
    float4 v = src[i];
    float f[4] = {v.x, v.y, v.z, v.w};
    #pragma unroll
    for (int c = 0; c < 4; ++c) {
      uint k = fkey(f[c]);
      if ((k >> 10) >= thr) {
        uint pos = atomicAdd(&ccnt[b], 1u);
        if (pos < MAXCAND) {
          ckey[(size_t)b * MAXCAND + pos] = k;
          cidx[(size_t)b * MAXCAND + pos] = (uint)(i * 4 + c);
        }
      }
    }
  }
}

// ---- bitonic sort of (key<<32 | ~idx), descending -> exact jax top_k order ----
__global__ void k_sort(const uint* __restrict__ ccnt, const uint* __restrict__ ckey,
                       const uint* __restrict__ cidx, float* __restrict__ tsc,
                       uint* __restrict__ tix) {
  const int b = blockIdx.x;
  const int tid = threadIdx.x;
  __shared__ ull sk[MAXCAND];
  uint cnt = ccnt[b]; if (cnt > MAXCAND) cnt = MAXCAND;
  for (int i = tid; i < MAXCAND; i += 1024)
    sk[i] = (i < (int)cnt)
      ? (((ull)ckey[(size_t)b * MAXCAND + i] << 32) | (uint)(~cidx[(size_t)b * MAXCAND + i]))
      : 0ull;
  for (uint size = 2; size <= MAXCAND; size <<= 1) {
    for (uint str = size >> 1; str > 0; str >>= 1) {
      __syncthreads();
      for (int i = tid; i < MAXCAND; i += 1024) {
        int j = i ^ (int)str;
        if (j > i) {
          bool desc = ((i & size) == 0);
          ull x = sk[i], y = sk[j];
          if (desc ? (x < y) : (x > y)) { sk[i] = y; sk[j] = x; }
        }
      }
    }
  }
  __syncthreads();
  if (tid < TOPK) {
    ull e = sk[tid];
    tsc[b * TOPK + tid] = fkey_inv((uint)(e >> 32));
    tix[b * TOPK + tid] = ~(uint)(e & 0xFFFFFFFFu);
  }
}

__global__ void k_decode(const float* __restrict__ reg, const float* __restrict__ anc,
                         const float* __restrict__ tsc, const uint* __restrict__ tix,
                         float* __restrict__ box, float* __restrict__ vol,
                         uint* __restrict__ valid, int N) {
  const int g = blockIdx.x * blockDim.x + threadIdx.x;
  if (g >= NB * KP) return;
  const int b = g / KP, i = g % KP;
  float bx[6] = {0.f, 0.f, 0.f, 0.f, 0.f, 0.f};
  float vv = 0.f; uint vd = 0u;
  if (i < TOPK) {
    uint idx = tix[b * TOPK + i];
    if (idx < (uint)N) {
      const float* rg = reg + ((size_t)b * N + idx) * 6;
      const float* an = anc + (size_t)idx * 6;
      float s = tsc[b * TOPK + i];
      #pragma unroll
      for (int d = 0; d < 3; ++d) {
        float ctr = rg[d] * an[3 + d] + an[d];
        float whd = expf(rg[3 + d]) * an[3 + d];
        bx[d]     = fmaxf(ctr - 0.5f * whd, 0.f);
        bx[3 + d] = fminf(ctr + 0.5f * whd, 127.f);
      }
      vv = (bx[3] - bx[0]) * (bx[4] - bx[1]) * (bx[5] - bx[2]);
      vd = (s > MIN_SCORE_C && vv > MIN_VOL_C) ? 1u : 0u;
    }
  }
  #pragma unroll
  for (int d = 0; d < 6; ++d) box[(size_t)g * 6 + d] = bx[d];
  vol[g] = vv;
  valid[g] = vd;
}

// ---- suppression mask: one wave per 16x16 tile; union volumes via WMMA ----
__global__ void k_mask(const float* __restrict__ box, const float* __restrict__ vol,
                       uint* __restrict__ smask) {
  const int tj = blockIdx.x, ti = blockIdx.y, b = blockIdx.z;
  if (tj < ti) return;                       // all j < i in these tiles
  const int lane = threadIdx.x;
  const int i0 = ti * 16, j0 = tj * 16;
  const float* volb = vol + (size_t)b * KP;
  const float* boxb = box + (size_t)b * KP * 6;

  // D[m][n] = vol_i[i0+m] + vol_j[j0+n] via [v,1,0,0] x [[1..1],[v],[0],[0]]
  v2f a, bb;
  a.x  = (lane < 16) ? volb[i0 + lane] : 0.f;   // A: lanes 0-15 give K=0,1
  a.y  = (lane < 16) ? 1.f : 0.f;
  bb.x = (lane < 16) ? 1.f : 0.f;               // B: lanes 0-15 give rows K=0,1
  bb.y = (lane < 16) ? volb[j0 + lane] : 0.f;
  v8f c = {0.f, 0.f, 0.f, 0.f, 0.f, 0.f, 0.f, 0.f};
  c = __builtin_amdgcn_wmma_f32_16x16x4_f32(false, a, false, bb, (short)0, c, false, false);

  const int col = j0 + (lane & 15);
  const int rbase = i0 + ((lane >> 4) << 3);    // lanes>=16 hold rows M=r+8
  const float bj0 = boxb[col * 6 + 0], bj1 = boxb[col * 6 + 1], bj2 = boxb[col * 6 + 2];
  const float bj3 = boxb[col * 6 + 3], bj4 = boxb[col * 6 + 4], bj5 = boxb[col * 6 + 5];
  uint* mrow = smask + (size_t)b * KP * 32 + (j0 >> 5);
  const uint sh = (uint)(j0 & 31);
  #pragma unroll
  for (int r = 0; r < 8; ++r) {
    const int row = rbase + r;
    const float* bi = boxb + (size_t)row * 6;
    float dx = fminf(bi[3], bj3) - fmaxf(bi[0], bj0);
    float dy = fminf(bi[4], bj4) - fmaxf(bi[1], bj1);
    float dz = fminf(bi[5], bj5) - fmaxf(bi[2], bj2);
    float inter = fmaxf(dx, 0.f) * fmaxf(dy, 0.f) * fmaxf(dz, 0.f);
    float un = c[r] - inter;                    // vol_i + vol_j - inter
    bool pred = (inter > 0.f) && (inter >= NMS_THR_C * un) && (col > row);
    uint bal = (uint)__ballot(pred);            // wave32: low 32 bits
    if (lane == 0)  atomicOr(mrow + (size_t)(i0 + r) * 32, (bal & 0xFFFFu) << sh);
    if (lane == 16) atomicOr(mrow + (size_t)(i0 + 8 + r) * 32, ((bal >> 16) & 0xFFFFu) << sh);
  }
}

// ---- sequential NMS scan (one wave per image) + pack outputs ----
__global__ void k_nms(const float* __restrict__ tsc, const float* __restrict__ box,
                      const uint* __restrict__ valid, const uint* __restrict__ smask,
                      float* __restrict__ outS, float* __restrict__ outB) {
  const int b = blockIdx.x;
  const int lane = threadIdx.x;
  uint kw = 0u;                                  // keep bits for i in [lane*32, lane*32+32)
  for (int t = 0; t < 32; ++t) {
    int i = lane * 32 + t;
    if (i < TOPK && valid[(size_t)b * KP + i]) kw |= (1u << t);
  }
  const uint* mb = smask + (size_t)b * KP * 32;
  for (int i = 0; i < TOPK; ++i) {
    uint wbits = (uint)__shfl((int)kw, i >> 5, 32);
    if ((wbits >> (i & 31)) & 1u)
      kw &= ~mb[(size_t)i * 32 + lane];
  }
  __shared__ uint kl[32];
  __shared__ uint pfx[32];
  kl[lane] = kw;
  __syncthreads();
  uint p = 0;
  for (int w = 0; w < lane; ++w) p += __popc(kl[w]);
  pfx[lane] = p;
  __syncthreads();
  for (int k = lane; k < MAXDET; k += 32) {
    outS[b * MAXDET + k] = -1.f;
    #pragma unroll
    for (int d = 0; d < 6; ++d) outB[((size_t)b * MAXDET + k) * 6 + d] = -1.f;
  }
  __syncthreads();
  const float* boxb = box + (size_t)b * KP * 6;
  for (int i = lane; i < TOPK; i += 32) {
    uint wv = kl[i >> 5];
    if ((wv >> (i & 31)) & 1u) {
      uint rank = pfx[i >> 5] + __popc(wv & ((1u << (i & 31)) - 1u));
      if (rank < MAXDET) {
        outS[b * MAXDET + rank] = tsc[b * TOPK + i];
        #pragma unroll
        for (int d = 0; d < 6; ++d)
          outB[((size_t)b * MAXDET + rank) * 6 + d] = boxb[(size_t)i * 6 + d];
      }
    }
  }
}

extern "C" void kernel_launch(void* const* d_in, const int* in_sizes, int n_in,
                              void* d_out, int out_size, void* d_ws, size_t ws_size,
                              hipStream_t stream) {
  const float* cls = (const float*)d_in[0];
  const float* reg = (const float*)d_in[1];
  const float* anc = (const float*)d_in[2];
  float* out = (float*)d_out;
  uint* ws = (uint*)d_ws;
  const int N = in_sizes[0] / NB;

  uint* hist1 = ws + OFF_HIST1;
  uint* hist2 = ws + OFF_HIST2;
  uint* ccnt  = ws + OFF_CCNT;
  uint* ckey  = ws + OFF_CKEY;
  uint* smask = ws + OFF_SMASK;
  uint* cidx  = ws + OFF_CIDX;
  uint* sel   = ws + OFF_SEL;
  float* tsc  = (float*)(ws + OFF_TSC);
  uint* tix   = ws + OFF_TIX;
  float* box  = (float*)(ws + OFF_BOX);
  float* vol  = (float*)(ws + OFF_VOL);
  uint* valid = ws + OFF_VALID;

  k_zero<<<512, 256, 0, stream>>>(ws, (int)ZWORDS);
  k_hist1<<<dim3(256, NB), 256, 0, stream>>>(cls, hist1, N);
  k_select1<<<NB, 32, 0, stream>>>(hist1, sel);
  k_hist2<<<dim3(256, NB), 256, 0, stream>>>(cls, hist2, sel, N);
  k_select2<<<NB, 32, 0, stream>>>(hist2, sel);
  k_compact<<<dim3(256, NB), 256, 0, stream>>>(cls, sel, ccnt, ckey, cidx, N);
  k_sort<<<NB, 1024, 0, stream>>>(ccnt, ckey, cidx, tsc, tix);
  k_decode<<<(NB * KP + 255) / 256, 256, 0, stream>>>(reg, anc, tsc, tix, box, vol, valid, N);
  k_mask<<<dim3(NTILE, NTILE, NB), 32, 0, stream>>>(box, vol, smask);
  k_nms<<<NB, 32, 0, stream>>>(tsc, box, valid, smask, out, out + NB * MAXDET);
}